// TritonDualLIF_50955492000360
// MI455X (gfx1250) — compile-verified
//
#include <hip/hip_runtime.h>

typedef __attribute__((ext_vector_type(2))) float v2f;
typedef __attribute__((ext_vector_type(8))) float v8f;

// Problem constants (from the reference)
constexpr int T_ = 16;
constexpr int B_ = 32;
constexpr int N_ = 196;
constexpr int C_ = 512;

constexpr int CT  = 64;          // channels per block
constexpr int NG  = 4;           // n-groups per block (thread rows)
constexpr int NI  = 49;          // n iterations per thread (N = NG*NI)
constexpr int DEPTH = 32;        // async ring depth (chunks of 1KB)
constexpr int AHEAD = 30;        // chunks issued ahead of consumption
constexpr int CHUNK_ELEMS = NG * CT;   // 256 floats = 1KB per chunk

// One 4-byte async global->LDS element per lane (ASYNCcnt-tracked).
__device__ __forceinline__ void async_ld_b32(unsigned lds_off, const float* g) {
  unsigned long long ga = (unsigned long long)(const void*)g;
  asm volatile("global_load_async_to_lds_b32 %0, %1, off"
               :: "v"(lds_off), "v"(ga) : "memory");
}
__device__ __forceinline__ void wait_async_le30() {
  asm volatile("s_wait_asynccnt 0x1e" ::: "memory");
}

__global__ __launch_bounds__(256) void lif_dual_kernel(
    const float* __restrict__ x, const float* __restrict__ decay,
    float* __restrict__ out) {
  __shared__ float ring[DEPTH][CHUNK_ELEMS];   // 32 KB async staging ring
  __shared__ float pacc[NG * CT];              // per-(n_grp, c) partial sums

  const int tid  = threadIdx.x;
  const int cidx = tid & (CT - 1);   // 0..63
  const int ng   = tid >> 6;         // 0..3
  const int blk  = blockIdx.x;
  const int ctile = blk & 7;         // C/CT = 8
  const int b     = blk >> 3;
  const int c0    = ctile * CT;

  const float dcy = decay[0];

  // Per-thread element base at t=0: x[((0*B+b)*N + ng)*C + c0 + cidx]
  const float* base = x + ((size_t)b * N_ + ng) * (size_t)C_ + c0 + cidx;
  const size_t tstride = (size_t)B_ * N_ * C_;

  // LDS byte offset of this thread's slot-0 element (low 32 bits of generic ptr)
  const unsigned myLds =
      (unsigned)(unsigned long long)(const void*)&ring[0][ng * CT + cidx];

  // Prologue: issue chunks (t=0, i=0..AHEAD-1) into slots 0..AHEAD-1
  #pragma unroll
  for (int i = 0; i < AHEAD; ++i) {
    async_ld_b32(myLds + (unsigned)(i * CHUNK_ELEMS * 4),
                 base + (size_t)(4 * i) * C_);
  }

  float v[NI];
  #pragma unroll
  for (int i = 0; i < NI; ++i) v[i] = 0.0f;

  float* spk  = out;                                    // [T,B,N,C]
  float* pool = out + (size_t)T_ * B_ * N_ * C_;        // [T,B,C]

  for (int t = 0; t < T_; ++t) {
    const float* baseT = base + (size_t)t * tstride;
    float* spkT = spk + ((size_t)(t * B_ + b) * N_ + ng) * (size_t)C_ + c0 + cidx;
    const int kb = t * NI;   // global chunk index base for this t
    float acc = 0.0f;

    #pragma unroll
    for (int i = 0; i < NI; ++i) {
      // Issue chunk kb+i+AHEAD (clamped to the final chunk at the tail).
      {
        const int ip = i + AHEAD;                 // compile-time per body
        const float* p;
        if (ip < NI) {
          p = baseT + (size_t)(4 * ip) * C_;
        } else if (t < T_ - 1) {
          p = baseT + tstride + (size_t)(4 * (ip - NI)) * C_;
        } else {
          p = baseT + (size_t)(4 * (NI - 1)) * C_;   // harmless re-issue
        }
        const unsigned slot = (unsigned)(kb + i + AHEAD) & (DEPTH - 1);
        async_ld_b32(myLds + slot * (CHUNK_ELEMS * 4), p);
      }
      wait_async_le30();   // oldest (= chunk kb+i) complete

      const float xv = ring[(kb + i) & (DEPTH - 1)][ng * CT + cidx];
      const float vv = fmaf(dcy, v[i], xv);   // charge
      acc += vv;                              // pre-reset pool contribution
      const float s = (vv >= 1.0f) ? 1.0f : 0.0f;
      v[i] = (vv >= 1.0f) ? 0.0f : vv;        // hard reset
      __builtin_nontemporal_store(s, spkT + (size_t)(4 * i) * C_);
    }

    // Reduce the 4 n-group partials per channel with an exact f32 WMMA:
    // D = Ones(16x4) x P(4x16) -> every row of D = per-channel sums.
    pacc[ng * CT + cidx] = acc;
    __syncthreads();

    const int w = tid >> 5;          // wave id (uniform per wave)
    if (w < 4) {                     // 4 waves cover 4 x 16 channels
      const int lane = tid & 31;
      const int half = lane >> 4;    // 0: rows {0,1}, 1: rows {2,3}
      const int l    = lane & 15;
      const int cc   = (w << 4) + l;
      v2f a;  a.x = 1.0f; a.y = 1.0f;              // ones A-matrix
      v2f bm;                                      // B 4x16 lane layout
      bm.x = pacc[(half * 2 + 0) * CT + cc];       // K = 0 / 2
      bm.y = pacc[(half * 2 + 1) * CT + cc];       // K = 1 / 3
      v8f cz = {0.f, 0.f, 0.f, 0.f, 0.f, 0.f, 0.f, 0.f};
      v8f dmat = __builtin_amdgcn_wmma_f32_16x16x4_f32(
          false, a, false, bm, (short)0, cz, false, false);
      if (half == 0) {   // row 0 (VGPR0, lanes 0..15) holds the column sums
        pool[((size_t)t * B_ + b) * C_ + c0 + cc] = dmat[0] * (1.0f / 196.0f);
      }
    }
    __syncthreads();   // pacc reused next t
  }
}

extern "C" void kernel_launch(void* const* d_in, const int* in_sizes, int n_in,
                              void* d_out, int out_size, void* d_ws, size_t ws_size,
                              hipStream_t stream) {
  const float* x     = (const float*)d_in[0];
  const float* decay = (const float*)d_in[1];
  float* out = (float*)d_out;
  dim3 grid(B_ * (C_ / CT));   // 256 blocks
  dim3 block(256);             // 8 waves (wave32)
  lif_dual_kernel<<<grid, block, 0, stream>>>(x, decay, out);
}